// IntraConv_single_5205500362912
// MI455X (gfx1250) — compile-verified
//
#include <hip/hip_runtime.h>
#include <hip/hip_bf16.h>

typedef float v2f __attribute__((ext_vector_type(2)));
typedef float v8f __attribute__((ext_vector_type(8)));

#define D        128
#define KTOT     256           // fused K: [x | neigh_norm]
#define WSTRIDE  258           // padded LDS row stride (floats) -> conflict-free ds reads
#define EPS      1e-8f

// ---------------- zero workspace ----------------
__global__ void k_zero4(float4* p, int count4) {
    int i = blockIdx.x * blockDim.x + threadIdx.x;
    if (i < count4) p[i] = make_float4(0.f, 0.f, 0.f, 0.f);
}

// ---------------- edge scatter: neigh[dst]+=x[src]*w, deg[dst]+=w ----------------
// one wave32 per edge; lane handles 4 consecutive features (32*4 = 128)
__global__ void k_scatter(const float* __restrict__ x,
                          const int*   __restrict__ src,
                          const int*   __restrict__ dst,
                          const float* __restrict__ ew,
                          float* __restrict__ neigh,
                          float* __restrict__ deg,
                          int E) {
    int wave = (blockIdx.x * blockDim.x + threadIdx.x) >> 5;
    int lane = threadIdx.x & 31;
    if (wave >= E) return;
    int s = src[wave];
    int d = dst[wave];
    float w = ew[wave];
    float4 v = *(const float4*)(x + (size_t)s * D + lane * 4);
    float* nd = neigh + (size_t)d * D + lane * 4;
    atomicAdd(nd + 0, v.x * w);
    atomicAdd(nd + 1, v.y * w);
    atomicAdd(nd + 2, v.z * w);
    atomicAdd(nd + 3, v.w * w);
    if (lane == 0) atomicAdd(deg + d, w);
}

// ---------------- fused GEMM via V_WMMA_F32_16X16X4_F32 ----------------
// out[r][c] = sum_k x[r][k]*Wself[c][k] + sum_k (neigh[r][k]/(deg[r]+eps))*Wneigh[c][k]
//           + bself[c] + bias[c]
// LDS holds Wt[n][k] (n = out col 0..127, k = 0..255), row stride WSTRIDE.
__global__ void k_gemm(const float* __restrict__ x,
                       const float* __restrict__ neigh,
                       const float* __restrict__ deg,
                       const float* __restrict__ Wself,
                       const float* __restrict__ Wneigh,
                       const float* __restrict__ bself,
                       const float* __restrict__ bias,
                       float* __restrict__ out,
                       int n_tiles) {
    extern __shared__ float lds[];   // 128 * WSTRIDE floats

    // stage combined transposed weight as float2 pairs (8B-aligned in LDS)
    for (int i = threadIdx.x; i < (D * KTOT) / 2; i += blockDim.x) {
        int n = i >> 7;              // 128 float2 per output column
        int k = (i & 127) << 1;
        v2f v;
        if (k < D) v = *(const v2f*)(Wself  + (size_t)n * D + k);
        else       v = *(const v2f*)(Wneigh + (size_t)n * D + (k - D));
        *(v2f*)&lds[n * WSTRIDE + k] = v;
    }
    __syncthreads();

    int wid    = (blockIdx.x * blockDim.x + threadIdx.x) >> 5;
    int nwaves = (gridDim.x * blockDim.x) >> 5;
    int lane   = threadIdx.x & 31;
    int m      = lane & 15;          // row-in-tile for A, col-in-tile for B/C/D
    int koff   = (lane >> 4) << 1;   // A/B K-pair offset: lanes 0-15 -> 0, 16-31 -> 2

    // per-lane output-column bias (independent of tile)
    float bcol[8];
#pragma unroll
    for (int ct = 0; ct < 8; ++ct) bcol[ct] = bself[ct * 16 + m] + bias[ct * 16 + m];

    for (int tile = wid; tile < n_tiles; tile += nwaves) {
        int row = tile * 16 + m;
        const float* xr = x     + (size_t)row * D + koff;
        const float* nr = neigh + (size_t)row * D + koff;
        const float* lb = lds + m * WSTRIDE + koff;   // B base for this lane (ct adds 16*WSTRIDE)
        float rinv = 1.0f / (deg[row] + EPS);

        v8f acc[8] = {};

        // ---- first half: A = x (straight-line, no branches) ----
#pragma unroll 4
        for (int k0 = 0; k0 < D; k0 += 4) {
            v2f a = *(const v2f*)(xr + k0);
#pragma unroll
            for (int ct = 0; ct < 8; ++ct) {
                v2f b = *(const v2f*)(lb + ct * 16 * WSTRIDE + k0);
                acc[ct] = __builtin_amdgcn_wmma_f32_16x16x4_f32(
                    false, a, false, b, (short)0, acc[ct], false, false);
            }
        }
        // ---- second half: A = neigh * (1/(deg+eps)) ----
#pragma unroll 4
        for (int k0 = 0; k0 < D; k0 += 4) {
            v2f t = *(const v2f*)(nr + k0);
            v2f a = t * rinv;
#pragma unroll
            for (int ct = 0; ct < 8; ++ct) {
                v2f b = *(const v2f*)(lb + ct * 16 * WSTRIDE + D + k0);
                acc[ct] = __builtin_amdgcn_wmma_f32_16x16x4_f32(
                    false, a, false, b, (short)0, acc[ct], false, false);
            }
        }

        // C/D layout: VGPR v -> M = v (lanes 0-15) or v+8 (lanes 16-31), N = lane%16
        int rbase = tile * 16 + ((lane >> 4) << 3);
#pragma unroll
        for (int v = 0; v < 8; ++v) {
            float* orow = out + (size_t)(rbase + v) * D;
#pragma unroll
            for (int ct = 0; ct < 8; ++ct) {
                orow[ct * 16 + m] = acc[ct][v] + bcol[ct];
            }
        }
    }
}

extern "C" void kernel_launch(void* const* d_in, const int* in_sizes, int n_in,
                              void* d_out, int out_size, void* d_ws, size_t ws_size,
                              hipStream_t stream) {
    const float* x      = (const float*)d_in[0];
    const int*   src    = (const int*)  d_in[1];
    const int*   dst    = (const int*)  d_in[2];
    const float* ew     = (const float*)d_in[3];
    const float* Wself  = (const float*)d_in[4];
    const float* bself  = (const float*)d_in[5];
    const float* Wneigh = (const float*)d_in[6];
    const float* bias   = (const float*)d_in[7];
    float*       out    = (float*)d_out;

    const int E = in_sizes[1];
    const int N = in_sizes[0] / D;

    float* neigh = (float*)d_ws;          // N*D floats
    float* deg   = neigh + (size_t)N * D; // N floats

    // 1) zero accumulators
    int count4 = (N * (D + 1)) / 4;       // N*129 floats, divisible by 4
    k_zero4<<<(count4 + 255) / 256, 256, 0, stream>>>((float4*)d_ws, count4);

    // 2) edge scatter: one wave per edge, 8 edges per 256-thread block
    int sblocks = (E + 7) / 8;
    k_scatter<<<sblocks, 256, 0, stream>>>(x, src, dst, ew, neigh, deg, E);

    // 3) fused dual-GEMM with f32 WMMA
    int n_tiles = N / 16;                 // 50000/16 = 3125, exact
    int gblocks = (n_tiles + 7) / 8;      // 8 waves per block, ~1 tile per wave
    size_t lds_bytes = (size_t)D * WSTRIDE * sizeof(float);  // ~129 KB < 320 KB/WGP
    k_gemm<<<gblocks, 256, lds_bytes, stream>>>(x, neigh, deg, Wself, Wneigh,
                                                bself, bias, out, n_tiles);
}